// NodeModel_11536282157154
// MI455X (gfx1250) — compile-verified
//
#include <hip/hip_runtime.h>

typedef __attribute__((ext_vector_type(2))) float v2f;
typedef __attribute__((ext_vector_type(8))) float v8f;
typedef __attribute__((ext_vector_type(4))) int   v4i;

// One workgroup per batch. 256 threads = 8 wave32.
// out[b,:] = sum_k w[k] * E[b, idx[k], :], w = mask/count, both events fused.
// Weighted row-sum via V_WMMA_F32_16X16X4_F32 (exact f32 FMA):
//   A (16x4): all 16 M-rows broadcast weights w[k..k+3]
//             (lanes 0-15 carry K={0,1}, lanes 16-31 carry K={2,3})
//   B (4x16): 4 gathered row fragments x 16 columns (same K split assumed)
//   C: accumulator; every M-row identical == the answer.
#define THREADS 256
#define WAVES 8
#define LMAX 64
#define MAXCHUNK 32   // 128 rows / 4 K per WMMA

__global__ void __launch_bounds__(THREADS)
masked_mean_gather_wmma(const float* __restrict__ embed,
                        const int* __restrict__ ev1,
                        const int* __restrict__ m1,
                        const int* __restrict__ ev2,
                        const int* __restrict__ m2,
                        float* __restrict__ out,
                        int N, int D, int L) {
    __shared__ int s_idx[2 * LMAX];
    __shared__ int s_m[2 * LMAX];
    __shared__ int s_cnt[2];
    // Packed per-chunk records: [chunk][half] = {w0.bits, w1.bits, elemoff0, elemoff1}
    __shared__ int s_pack[MAXCHUNK * 2 * 4];
    __shared__ int s_nchunk;

    const int b   = blockIdx.x;
    const int tid = threadIdx.x;

    if (tid < 2) s_cnt[tid] = 0;
    __syncthreads();

    // Phase 1: stage indices + masks, count active per event.
    if (tid < L) {
        s_idx[tid] = ev1[b * L + tid];
        int m = m1[b * L + tid];
        s_m[tid] = m;
        if (m) atomicAdd(&s_cnt[0], 1);
    } else if (tid < 2 * L) {
        int l = tid - L;
        s_idx[tid] = ev2[b * L + l];
        int m = m2[b * L + l];
        s_m[tid] = m;
        if (m) atomicAdd(&s_cnt[1], 1);
    }
    __syncthreads();

    // Compact active rows into packed ds_load_b128-friendly records.
    // Pad to a multiple of 16 rows (4 chunks) with w=0, row 0 (harmless: cached, annihilated).
    if (tid == 0) {
        int   c1   = s_cnt[0] > 0 ? s_cnt[0] : 1;
        int   c2   = s_cnt[1] > 0 ? s_cnt[1] : 1;
        float inv1 = 1.0f / (float)c1;
        float inv2 = 1.0f / (float)c2;
        int n = 0;
        for (int l = 0; l < 2 * L; ++l) {
            if (s_m[l]) {
                float w  = (l < L) ? inv1 : inv2;
                int chunk = n >> 2, wi = n & 3, hf = wi >> 1, ps = wi & 1;
                int base  = (chunk * 2 + hf) * 4;
                s_pack[base + ps]     = __float_as_int(w);
                s_pack[base + 2 + ps] = s_idx[l] * D;   // element offset of row start
                ++n;
            }
        }
        while (n & 15) {
            int chunk = n >> 2, wi = n & 3, hf = wi >> 1, ps = wi & 1;
            int base  = (chunk * 2 + hf) * 4;
            s_pack[base + ps]     = 0;   // w = 0.0f
            s_pack[base + 2 + ps] = 0;   // row 0
            ++n;
        }
        s_nchunk = n >> 2;
    }
    __syncthreads();

    // Phase 2: WMMA-accumulated weighted gather; 2 column tiles per wave iteration
    // so each 16-lane group pulls a full 128B cacheline span per row.
    // Loads use uniform-base (saddr) + 32-bit element offsets so the compiler can
    // emit global_load ... s[base] scale_offset instead of 64-bit per-lane addressing.
    const float* __restrict__ Eb = embed + (size_t)b * (size_t)N * (size_t)D;
    const int lane = tid & 31;
    const int wave = tid >> 5;
    const int half = lane >> 4;     // A/B K-split: lanes 0-15 -> K={0,1}, 16-31 -> K={2,3}
    const int nh   = lane & 15;     // N position within tile
    const int nchunk = s_nchunk;
    __builtin_assume(nchunk >= 0 && nchunk <= MAXCHUNK && (nchunk & 3) == 0);
    const v4i* __restrict__ packs = (const v4i*)s_pack;

    const int ntiles = D >> 4;
    const int npairs = ntiles >> 1;

    for (int p = wave; p < npairs; p += WAVES) {
        const unsigned col0 = (unsigned)(p * 32 + nh);  // tiles (2p, 2p+1)
        v8f acc0 = {}, acc1 = {};
#pragma unroll 4
        for (int c = 0; c < nchunk; ++c) {
            v4i pk = packs[c * 2 + half];               // single ds_load_b128
            v2f a;  a[0] = __int_as_float(pk[0]); a[1] = __int_as_float(pk[1]);
            const unsigned u0 = (unsigned)pk[2] + col0; // 32-bit element offsets
            const unsigned u1 = (unsigned)pk[3] + col0;
            v2f b0; b0[0] = Eb[u0];      b0[1] = Eb[u1];
            v2f b1; b1[0] = Eb[u0 + 16]; b1[1] = Eb[u1 + 16];
            acc0 = __builtin_amdgcn_wmma_f32_16x16x4_f32(
                false, a, false, b0, (short)0, acc0, false, false);
            acc1 = __builtin_amdgcn_wmma_f32_16x16x4_f32(
                false, a, false, b1, (short)0, acc1, false, false);
        }
        if (lane < 16) {
            out[(size_t)b * (size_t)D + col0]      = acc0[0];
            out[(size_t)b * (size_t)D + col0 + 16] = acc1[0];
        }
    }

    // Odd leftover tile (not hit for D=768): wave-uniform condition keeps EXEC full.
    if ((ntiles & 1) && wave == 0) {
        const unsigned col = (unsigned)((ntiles - 1) * 16 + nh);
        v8f acc = {};
#pragma unroll 4
        for (int c = 0; c < nchunk; ++c) {
            v4i pk = packs[c * 2 + half];
            v2f a;  a[0] = __int_as_float(pk[0]); a[1] = __int_as_float(pk[1]);
            v2f bb; bb[0] = Eb[(unsigned)pk[2] + col]; bb[1] = Eb[(unsigned)pk[3] + col];
            acc = __builtin_amdgcn_wmma_f32_16x16x4_f32(
                false, a, false, bb, (short)0, acc, false, false);
        }
        if (lane < 16) out[(size_t)b * (size_t)D + col] = acc[0];
    }
}

extern "C" void kernel_launch(void* const* d_in, const int* in_sizes, int n_in,
                              void* d_out, int out_size, void* d_ws, size_t ws_size,
                              hipStream_t stream) {
    const float* embed = (const float*)d_in[0];
    const int*   ev1   = (const int*)d_in[1];
    const int*   m1    = (const int*)d_in[2];
    const int*   ev2   = (const int*)d_in[3];
    const int*   m2    = (const int*)d_in[4];
    float*       out   = (float*)d_out;

    const int L = 64;                                  // reference fixes L=64
    const int B = in_sizes[1] / L;                     // [B, L]
    const int D = out_size / B;                        // [B, D]
    const int N = (int)((long long)in_sizes[0] / ((long long)B * D)); // [B, N, D]

    masked_mean_gather_wmma<<<B, THREADS, 0, stream>>>(embed, ev1, m1, ev2, m2, out, N, D, L);
}